// MultiMambaBlock_43757126812046
// MI455X (gfx1250) — compile-verified
//
#include <hip/hip_runtime.h>
#include <hip/hip_bf16.h>
#include <math.h>
#include <stdint.h>

// ---------------------------------------------------------------------------
// MultiMambaBlock for MI455X (gfx1250, wave32).
//   B=2, N=4, L=512, D_MODEL=512, D_INNER=1024, DT_RANK=32, D_STATE=16, D_CONV=4
//   R = B*N*L = 4096 rows.
// Pipeline:
//   1) wmma_gemm      : x[4096,512] @ W_in[512,2048]   -> xr[4096,2048]
//   2) conv_silu      : depthwise causal conv + silu   -> u[4096,1024]
//   3) wmma_gemm      : u @ W_x[1024,64]               -> dtBC[4096,64]
//   4) wmma_gemm(+sp) : dtBC[:, :32] @ W_dt[32,1024] + b_dt, softplus -> dt
//   5) scan_kernel    : selective SSM scan + gating    -> g (in-place over u)
//   6) wmma_gemm      : g @ W_out[1024,512]            -> out[4096,512]
// GEMMs: V_WMMA_F32_16X16X4_F32 (exact f32) + double-buffered
// GLOBAL_LOAD_ASYNC_TO_LDS_B128 staging (ASYNCcnt-tracked, overlapped).
// ---------------------------------------------------------------------------

typedef __attribute__((ext_vector_type(2))) float v2f;
typedef __attribute__((ext_vector_type(8))) float v8f;

#define R_ROWS 4096
#define D_INNER 1024
#define D_STATE 16

__device__ __forceinline__ v8f wmma_f32_k4(v2f a, v2f b, v8f c) {
  return __builtin_amdgcn_wmma_f32_16x16x4_f32(
      /*neg_a=*/false, a, /*neg_b=*/false, b,
      /*c_mod=*/(short)0, c, /*reuse_a=*/false, /*reuse_b=*/false);
}

// low 32 bits of a generic pointer to __shared__ = LDS byte offset
__device__ __forceinline__ uint32_t lds_off(const void* p) {
  return (uint32_t)(uintptr_t)p;
}

// Async global->LDS 128b per lane (GVS mode: SGPR64 base + VGPR32 byte offset).
__device__ __forceinline__ void async_ld_b128(uint32_t lds, const float* sbase,
                                              uint32_t voff) {
  asm volatile("global_load_async_to_lds_b128 %0, %1, %2"
               :: "v"(lds), "v"(voff), "s"(sbase) : "memory");
}

// Generic row-major f32 GEMM: C[M,N] = A[M,K] @ B[K,N]  (+ optional softplus(x+bias))
// Block = 128 threads = 4 waves; block tile 32(M) x 64(N); K staged 16 at a time.
// Double-buffered async LDS staging: 3 async b128 per wave per chunk
// (1 for the 32x16 A tile, 2 for the 16x64 B tile).
__global__ __launch_bounds__(128)
void wmma_gemm(const float* __restrict__ A, const float* __restrict__ B,
               float* __restrict__ C, int M, int N, int K,
               int lda, int ldb, int ldc,
               const float* __restrict__ bias, int softplus_mode) {
  __shared__ float As[2][32][20];   // padded rows: conflict-free fragment reads
  __shared__ float Bs[2][16][64];

  const int tid  = threadIdx.x;
  const int lane = tid & 31;
  const int wave = tid >> 5;        // 0..3
  const int wm   = wave & 1;        // row-wave
  const int wn   = wave >> 1;       // col-wave
  const int row0 = blockIdx.y * 32;
  const int col0 = blockIdx.x * 64;

  const int m16 = lane & 15;        // matrix row (A) / col (B,C) within tile
  const int kb  = (lane >> 4) * 2;  // K-pair selector for A/B fragments

  // --- per-lane async staging coordinates ---
  // A tile 32x16 floats: wave w stages rows w*8..w*8+7, 4 lanes per row (16B each)
  const int aRow  = wave * 8 + (lane >> 2);
  const int aCol4 = (lane & 3) * 4;
  const uint32_t aLds0  = lds_off(&As[0][aRow][aCol4]);
  const uint32_t aVoff0 = (uint32_t)(((row0 + aRow) * lda + aCol4) * 4);
  // B tile 16x64 floats: 8 wave-loads of 2 rows each; wave w does q=w and q=w+4
  const int bCol4 = (lane & 15) * 4;
  const int bRowA = wave * 2 + (lane >> 4);        // rows 0..7
  const int bRowB = (wave + 4) * 2 + (lane >> 4);  // rows 8..15
  const uint32_t bLds0A = lds_off(&Bs[0][bRowA][bCol4]);
  const uint32_t bLds0B = lds_off(&Bs[0][bRowB][bCol4]);
  const uint32_t bVoffA = (uint32_t)((bRowA * ldb + col0 + bCol4) * 4);
  const uint32_t bVoffB = (uint32_t)((bRowB * ldb + col0 + bCol4) * 4);

  const uint32_t A_BUF = (uint32_t)sizeof(float) * 32u * 20u;  // 2560 B
  const uint32_t B_BUF = (uint32_t)sizeof(float) * 16u * 64u;  // 4096 B

  v8f acc0 = {0.f, 0.f, 0.f, 0.f, 0.f, 0.f, 0.f, 0.f};
  v8f acc1 = acc0;

  const int nk = K >> 4;
  int buf = 0;

  // prologue: stage chunk 0 into buffer 0
  async_ld_b128(aLds0, A, aVoff0);
  async_ld_b128(bLds0A, B, bVoffA);
  async_ld_b128(bLds0B, B, bVoffB);

  for (int kt = 0; kt < nk; ++kt) {
    if (kt + 1 < nk) {
      // stage next chunk into the other buffer, then drain only current chunk
      const int nb = buf ^ 1;
      const uint32_t k4  = (uint32_t)((kt + 1) << 4) * 4u;         // k0 * 4B
      const uint32_t kB4 = (uint32_t)(((kt + 1) << 4) * ldb) * 4u; // k0*ldb*4B
      async_ld_b128(aLds0 + (uint32_t)nb * A_BUF, A, aVoff0 + k4);
      async_ld_b128(bLds0A + (uint32_t)nb * B_BUF, B, bVoffA + kB4);
      async_ld_b128(bLds0B + (uint32_t)nb * B_BUF, B, bVoffB + kB4);
      asm volatile("s_wait_asynccnt 0x3" ::: "memory");
    } else {
      asm volatile("s_wait_asynccnt 0x0" ::: "memory");
    }
    __syncthreads();

    #pragma unroll
    for (int kk = 0; kk < 4; ++kk) {
      const int kbase = kk * 4 + kb;
      v2f a;
      a.x = As[buf][wm * 16 + m16][kbase];
      a.y = As[buf][wm * 16 + m16][kbase + 1];
      v2f b0, b1;
      b0.x = Bs[buf][kbase][wn * 32 + m16];
      b0.y = Bs[buf][kbase + 1][wn * 32 + m16];
      b1.x = Bs[buf][kbase][wn * 32 + 16 + m16];
      b1.y = Bs[buf][kbase + 1][wn * 32 + 16 + m16];
      acc0 = wmma_f32_k4(a, b0, acc0);
      acc1 = wmma_f32_k4(a, b1, acc1);
    }
    __syncthreads();   // all waves done reading buf before it is restaged
    buf ^= 1;
  }

  // --- store (C/D layout: VGPR j -> M = j + 8*(lane>=16), N = lane&15) ---
  const int hi = lane >> 4;
  const int cc0 = col0 + wn * 32 + m16;
  #pragma unroll
  for (int j = 0; j < 8; ++j) {
    const int m = row0 + wm * 16 + j + 8 * hi;
    float v0 = acc0[j];
    float v1 = acc1[j];
    if (softplus_mode) {
      v0 += bias[cc0];
      v1 += bias[cc0 + 16];
      v0 = (v0 > 20.f) ? v0 : log1pf(__expf(v0));
      v1 = (v1 > 20.f) ? v1 : log1pf(__expf(v1));
    }
    C[(size_t)m * ldc + cc0]      = v0;
    C[(size_t)m * ldc + cc0 + 16] = v1;
  }
}

// Depthwise causal conv (width 4, left pad 3) + bias + silu.
// xr rows r = ((b*4+n)*512 + l); channel c = n*1024 + d; x_in = xr[:, 0:1024].
__global__ __launch_bounds__(256)
void conv_silu_kernel(const float* __restrict__ xr, const float* __restrict__ cw,
                      const float* __restrict__ cb, float* __restrict__ u) {
  const int idx = blockIdx.x * 256 + threadIdx.x;   // 0 .. 4096*1024-1
  const int d = idx & (D_INNER - 1);
  const int r = idx >> 10;
  const int l = r & 511;
  const int n = (r >> 9) & 3;
  const int c = n * D_INNER + d;
  float acc = cb[c];
  #pragma unroll
  for (int j = 0; j < 4; ++j) {
    const int lj = l - 3 + j;
    if (lj >= 0) acc += xr[(size_t)(r - 3 + j) * 2048 + d] * cw[c * 4 + j];
  }
  const float sig = 1.f / (1.f + __expf(-acc));
  u[idx] = acc * sig;
}

// Selective scan: one thread per (seq bn, channel d); 16-state recurrence in regs.
// B/C vectors (dtBC cols 32..63) staged through LDS in 64-step chunks.
// Epilogue fuses y += u*D and gating by silu(res); writes g in place over u.
#define LCH 64
__global__ __launch_bounds__(256)
void scan_kernel(const float* __restrict__ dt, float* __restrict__ u,
                 const float* __restrict__ dtBC, const float* __restrict__ xr,
                 const float* __restrict__ A_log, const float* __restrict__ D_param) {
  const int bn   = blockIdx.x >> 2;           // 0..7
  const int dblk = blockIdx.x & 3;            // 0..3
  const int tid  = threadIdx.x;               // 0..255
  const int d    = dblk * 256 + tid;

  __shared__ float BC[LCH][32];               // [step][B(16) | C(16)]

  float h[D_STATE], Ar[D_STATE];
  #pragma unroll
  for (int s = 0; s < D_STATE; ++s) {
    h[s]  = 0.f;
    Ar[s] = -__expf(A_log[d * D_STATE + s]);
  }
  const float Dv = D_param[d];

  for (int l0 = 0; l0 < 512; l0 += LCH) {
    __syncthreads();
    for (int i = tid; i < LCH * 32; i += 256) {
      const int ll = i >> 5, j = i & 31;
      BC[ll][j] = dtBC[(size_t)(bn * 512 + l0 + ll) * 64 + 32 + j];
    }
    __syncthreads();

    // prefetch next chunk's streaming operands while this chunk computes
    if (l0 + LCH < 512) {
      const size_t nrow = (size_t)bn * 512 + l0 + LCH;
      __builtin_prefetch(&dt[nrow * D_INNER + d], 0, 0);
      __builtin_prefetch(&u[nrow * D_INNER + d], 0, 0);
      __builtin_prefetch(&xr[nrow * 2048 + D_INNER + d], 0, 0);
    }

    for (int li = 0; li < LCH; ++li) {
      const size_t row = (size_t)bn * 512 + l0 + li;
      const float dtv  = dt[row * D_INNER + d];
      const float uv   = u[row * D_INNER + d];
      const float resv = xr[row * 2048 + D_INNER + d];
      float y = 0.f;
      #pragma unroll
      for (int s = 0; s < D_STATE; ++s) {
        const float dA = __expf(dtv * Ar[s]);
        h[s] = dA * h[s] + (dtv * BC[li][s]) * uv;
        y += h[s] * BC[li][16 + s];
      }
      y += uv * Dv;
      const float sig = 1.f / (1.f + __expf(-resv));
      u[row * D_INNER + d] = y * (resv * sig);   // gated activation g
    }
  }
}

extern "C" void kernel_launch(void* const* d_in, const int* in_sizes, int n_in,
                              void* d_out, int out_size, void* d_ws, size_t ws_size,
                              hipStream_t stream) {
  (void)in_sizes; (void)n_in; (void)out_size; (void)ws_size;

  const float* x     = (const float*)d_in[0];
  const float* W_in  = (const float*)d_in[1];
  const float* convw = (const float*)d_in[2];
  const float* convb = (const float*)d_in[3];
  const float* W_x   = (const float*)d_in[4];
  const float* W_dt  = (const float*)d_in[5];
  const float* b_dt  = (const float*)d_in[6];
  const float* A_log = (const float*)d_in[7];
  const float* D_par = (const float*)d_in[8];
  const float* W_out = (const float*)d_in[9];
  float* out = (float*)d_out;

  float* xr   = (float*)d_ws;                       // [4096, 2048]
  float* ubuf = xr   + (size_t)R_ROWS * 2048;       // [4096, 1024] (u, then g)
  float* dtBC = ubuf + (size_t)R_ROWS * 1024;       // [4096, 64]
  float* dt   = dtBC + (size_t)R_ROWS * 64;         // [4096, 1024]

  const dim3 blk(128);

  // 1) in_proj: xr = x @ W_in
  wmma_gemm<<<dim3(2048 / 64, R_ROWS / 32), blk, 0, stream>>>(
      x, W_in, xr, R_ROWS, 2048, 512, 512, 2048, 2048, nullptr, 0);

  // 2) depthwise causal conv + silu -> u
  conv_silu_kernel<<<(R_ROWS * D_INNER) / 256, 256, 0, stream>>>(xr, convw, convb, ubuf);

  // 3) dtBC = u @ W_x
  wmma_gemm<<<dim3(64 / 64, R_ROWS / 32), blk, 0, stream>>>(
      ubuf, W_x, dtBC, R_ROWS, 64, 1024, 1024, 64, 64, nullptr, 0);

  // 4) dt = softplus(dtBC[:, :32] @ W_dt + b_dt)
  wmma_gemm<<<dim3(1024 / 64, R_ROWS / 32), blk, 0, stream>>>(
      dtBC, W_dt, dt, R_ROWS, 1024, 32, 64, 1024, 1024, b_dt, 1);

  // 5) selective scan + gating -> g (in place over u)
  scan_kernel<<<32, 256, 0, stream>>>(dt, ubuf, dtBC, xr, A_log, D_par);

  // 6) out = g @ W_out
  wmma_gemm<<<dim3(512 / 64, R_ROWS / 32), blk, 0, stream>>>(
      ubuf, W_out, out, R_ROWS, 512, 1024, 1024, 512, 512, nullptr, 0);
}